// TemporalGCN_850403524987
// MI455X (gfx1250) — compile-verified
//
#include <hip/hip_runtime.h>
#include <hip/hip_bf16.h>

typedef float v2f __attribute__((ext_vector_type(2)));
typedef float v8f __attribute__((ext_vector_type(8)));

#define D 128
#define SW_PITCH 136   // LDS row pitch (floats): 2*136 mod 64 = 16 -> lane-half bank windows disjoint
#define NEG_SLOPE 0.2f

// ---------------------------------------------------------------------------
// Fill a float buffer with a constant (grid-stride).
// ---------------------------------------------------------------------------
__global__ __launch_bounds__(256) void fill_f32(float* __restrict__ p, long n, float v) {
    long i = (long)blockIdx.x * blockDim.x + threadIdx.x;
    long stride = (long)gridDim.x * blockDim.x;
    for (; i < n; i += stride) p[i] = v;
}

// ---------------------------------------------------------------------------
// C[rows x 128] = X[rows x 128] @ W[128 x 128]   (fp32, exact, via WMMA)
//
// Block = 256 threads = 8 waves. Each wave owns a 16-row strip and all 128
// output columns (8 tiles of 16x16). W staged in LDS with padded pitch (136)
// to kill 2-way bank conflicts between lane halves. All 32 A-fragments
// (b64 each) are preloaded into registers so the K-loop never waits on VMEM.
//
// V_WMMA_F32_16X16X4_F32 operand layout (ISA 7.12.2):
//   A 16x4:  lane m = lane&15 (M), half h = lane>>4; VGPR j holds K = j + 2h
//   B 4x16:  lane n = lane&15 (N), half h;           VGPR j holds K = j + 2h
//   C/D 16x16: VGPR r -> M = r (lanes 0-15) / M = r+8 (lanes 16-31), N = lane&15
// ---------------------------------------------------------------------------
__global__ __launch_bounds__(256) void gemm128_wmma(const float* __restrict__ X,
                                                    const float* __restrict__ W,
                                                    float* __restrict__ C, int rows) {
    __shared__ float sW[D * SW_PITCH];

    // cooperative load of W into LDS (float4), padded pitch
    {
        const float4* W4 = (const float4*)W;
        for (int idx = threadIdx.x; idx < D * (D / 4); idx += 256) {
            const int r = idx >> 5;        // row 0..127
            const int c4 = idx & 31;       // float4 column 0..31
            *(float4*)(sW + r * SW_PITCH + c4 * 4) = W4[r * (D / 4) + c4];
        }
    }
    __syncthreads();

    const int wave = threadIdx.x >> 5;
    const int lane = threadIdx.x & 31;
    const int m = lane & 15;
    const int h = lane >> 4;           // 0 or 1 (lane half)

    const long row0 = ((long)blockIdx.x * 8 + wave) * 16;
    if (row0 >= rows) return;          // wave-uniform: EXEC stays all-ones
    const long row = row0 + m;

    // Preload all A fragments for this lane: step kk covers K = 4*kk .. 4*kk+3,
    // this lane holds elements 4*kk + 2h, 4*kk + 2h + 1  (8-byte aligned b64).
    const float* xrow = X + row * D + 2 * h;
    v2f afrag[32];
#pragma unroll
    for (int kk = 0; kk < 32; ++kk)
        afrag[kk] = *(const v2f*)(xrow + kk * 4);

    v8f acc[8];
#pragma unroll
    for (int t = 0; t < 8; ++t) acc[t] = (v8f)0.0f;

#pragma unroll 4
    for (int kk = 0; kk < 32; ++kk) {
        const int krow = (kk * 4 + 2 * h) * SW_PITCH;
#pragma unroll
        for (int t = 0; t < 8; ++t) {
            const int base = krow + t * 16 + m;
            v2f b;
            b.x = sW[base];                 // W[k+2h][col]
            b.y = sW[base + SW_PITCH];      // W[k+2h+1][col]
            acc[t] = __builtin_amdgcn_wmma_f32_16x16x4_f32(
                false, afrag[kk], false, b, (short)0, acc[t], false, false);
        }
    }

#pragma unroll
    for (int t = 0; t < 8; ++t) {
#pragma unroll
        for (int r = 0; r < 8; ++r) {
            const long mrow = row0 + r + 8 * h;
            C[mrow * D + t * 16 + m] = acc[t][r];
        }
    }
}

// ---------------------------------------------------------------------------
// Float atomic max via signed/unsigned int punning (buffer init'd to -inf).
// ---------------------------------------------------------------------------
__device__ inline void atomicMaxFloat(float* addr, float v) {
    if (v >= 0.0f) atomicMax((int*)addr, __float_as_int(v));
    else           atomicMin((unsigned int*)addr, __float_as_uint(v));
}

__device__ inline float lrelu(float x) {
    return x > 0.0f ? x : NEG_SLOPE * x;
}

// edge e in [0,E): s=ei[e], d=ei[E+e];  e in [E,Etot): self-loop s=d=e-E
__device__ inline void edge_sd(const int* __restrict__ ei, int e, int E, int& s, int& d) {
    if (e < E) { s = ei[e]; d = ei[E + e]; }
    else       { s = e - E; d = e - E; }
}

// ---------------------------------------------------------------------------
// Pass 1: per-edge attention logit + segment max.  One wave32 per edge,
// 4 features per lane (float4 gathers), wave shuffle reduction.
// ---------------------------------------------------------------------------
__global__ __launch_bounds__(256) void edge_logits(const float* __restrict__ xl,
                                                   const float* __restrict__ xr,
                                                   const float* __restrict__ att,
                                                   const int* __restrict__ ei,
                                                   int E, int Etot,
                                                   float* __restrict__ logit,
                                                   float* __restrict__ emax) {
    const int lane = threadIdx.x & 31;
    const long gwave = ((long)blockIdx.x * blockDim.x + threadIdx.x) >> 5;
    const long nwaves = ((long)gridDim.x * blockDim.x) >> 5;

    const float4 w = *(const float4*)(att + lane * 4);

    for (long e = gwave; e < Etot; e += nwaves) {
        int s, d;
        edge_sd(ei, (int)e, E, s, d);
        const float4 a = *(const float4*)(xl + (long)s * D + lane * 4);
        const float4 b = *(const float4*)(xr + (long)d * D + lane * 4);
        float v = lrelu(a.x + b.x) * w.x
                + lrelu(a.y + b.y) * w.y
                + lrelu(a.z + b.z) * w.z
                + lrelu(a.w + b.w) * w.w;
#pragma unroll
        for (int off = 16; off >= 1; off >>= 1) v += __shfl_xor(v, off, 32);
        if (lane == 0) {
            logit[e] = v;
            atomicMaxFloat(emax + d, v);
        }
    }
}

// ---------------------------------------------------------------------------
// Pass 2: w = exp(logit - emax[d]); denom[d] += w.  One thread per edge.
// ---------------------------------------------------------------------------
__global__ __launch_bounds__(256) void edge_weight(const int* __restrict__ ei,
                                                   int E, int Etot,
                                                   const float* __restrict__ emax,
                                                   float* __restrict__ ew,
                                                   float* __restrict__ denom) {
    const long i0 = (long)blockIdx.x * blockDim.x + threadIdx.x;
    const long stride = (long)gridDim.x * blockDim.x;
    for (long e = i0; e < Etot; e += stride) {
        int s, d;
        edge_sd(ei, (int)e, E, s, d);
        (void)s;
        const float w = __expf(ew[e] - emax[d]);   // ew currently holds logits
        ew[e] = w;
        atomicAdd(denom + d, w);
    }
}

// ---------------------------------------------------------------------------
// Pass 3: out[d] += (w/denom[d]) * xl[s].  One wave32 per edge, float4 per
// lane, 4x global_atomic_add_f32 per lane.
// ---------------------------------------------------------------------------
__global__ __launch_bounds__(256) void edge_scatter(const float* __restrict__ xl,
                                                    const int* __restrict__ ei,
                                                    int E, int Etot,
                                                    const float* __restrict__ ew,
                                                    const float* __restrict__ denom,
                                                    float* __restrict__ out) {
    const int lane = threadIdx.x & 31;
    const long gwave = ((long)blockIdx.x * blockDim.x + threadIdx.x) >> 5;
    const long nwaves = ((long)gridDim.x * blockDim.x) >> 5;

    for (long e = gwave; e < Etot; e += nwaves) {
        int s, d;
        edge_sd(ei, (int)e, E, s, d);
        const float alpha = ew[e] / denom[d];
        const float4 a = *(const float4*)(xl + (long)s * D + lane * 4);
        float* o = out + (long)d * D + lane * 4;
        atomicAdd(o + 0, alpha * a.x);
        atomicAdd(o + 1, alpha * a.y);
        atomicAdd(o + 2, alpha * a.z);
        atomicAdd(o + 3, alpha * a.w);
    }
}

// ---------------------------------------------------------------------------
// Finalize: out = relu(acc + bias)  (may be in place)
// ---------------------------------------------------------------------------
__global__ __launch_bounds__(256) void finalize_relu_bias(const float* __restrict__ acc,
                                                          const float* __restrict__ bias,
                                                          float* __restrict__ out, long n) {
    const long i0 = (long)blockIdx.x * blockDim.x + threadIdx.x;
    const long stride = (long)gridDim.x * blockDim.x;
    for (long i = i0; i < n; i += stride) {
        const float v = acc[i] + bias[i & (D - 1)];
        out[i] = v > 0.0f ? v : 0.0f;
    }
}

// ---------------------------------------------------------------------------
// One GATv2 layer
// ---------------------------------------------------------------------------
static void run_layer(const float* hin, const float* Wl, const float* Wr,
                      const float* att, const float* bias,
                      float* xl, float* xr, float* acc, float* hout,
                      float* ew, float* emax, float* denom,
                      const int* ei, int N, int E, int Etot, hipStream_t stream) {
    const int tiles = (N + 15) / 16;
    const int gblocks = (tiles + 7) / 8;
    gemm128_wmma<<<gblocks, 256, 0, stream>>>(hin, Wl, xl, N);
    gemm128_wmma<<<gblocks, 256, 0, stream>>>(hin, Wr, xr, N);

    const long nElems = (long)N * D;
    fill_f32<<<(int)((nElems + 255) / 256), 256, 0, stream>>>(acc, nElems, 0.0f);
    fill_f32<<<(N + 255) / 256, 256, 0, stream>>>(emax, (long)N, -__builtin_huge_valf());
    fill_f32<<<(N + 255) / 256, 256, 0, stream>>>(denom, (long)N, 0.0f);

    const int wblocks = (Etot + 7) / 8;    // wave-per-edge kernels (8 waves/block)
    edge_logits<<<wblocks, 256, 0, stream>>>(xl, xr, att, ei, E, Etot, ew, emax);
    edge_weight<<<(Etot + 255) / 256, 256, 0, stream>>>(ei, E, Etot, emax, ew, denom);
    edge_scatter<<<wblocks, 256, 0, stream>>>(xl, ei, E, Etot, ew, denom, acc);

    finalize_relu_bias<<<(int)((nElems + 255) / 256), 256, 0, stream>>>(acc, bias, hout, nElems);
}

extern "C" void kernel_launch(void* const* d_in, const int* in_sizes, int n_in,
                              void* d_out, int out_size, void* d_ws, size_t ws_size,
                              hipStream_t stream) {
    const float* x    = (const float*)d_in[0];
    const int*   ei   = (const int*)d_in[1];
    const float* Wl1  = (const float*)d_in[2];
    const float* Wr1  = (const float*)d_in[3];
    const float* att1 = (const float*)d_in[4];
    const float* b1   = (const float*)d_in[5];
    const float* Wl2  = (const float*)d_in[6];
    const float* Wr2  = (const float*)d_in[7];
    const float* att2 = (const float*)d_in[8];
    const float* b2   = (const float*)d_in[9];

    const int N = in_sizes[0] / D;
    const int E = in_sizes[1] / 2;
    const int Etot = E + N;

    // workspace layout
    float* A     = (float*)d_ws;                 // xl, N*128
    float* B     = A + (size_t)N * D;            // xr, N*128
    float* H     = B + (size_t)N * D;            // hidden, N*128
    float* ew    = H + (size_t)N * D;            // per-edge logit/weight, Etot
    float* emax  = ew + (size_t)Etot;            // N
    float* denom = emax + (size_t)N;             // N

    float* out = (float*)d_out;

    // Layer 1: x -> H
    run_layer(x, Wl1, Wr1, att1, b1, A, B, H, H, ew, emax, denom, ei, N, E, Etot, stream);
    // Layer 2: H -> out  (accumulate + finalize directly in d_out)
    run_layer(H, Wl2, Wr2, att2, b2, A, B, out, out, ew, emax, denom, ei, N, E, Etot, stream);
}